// SIMCLR_5145370821233
// MI455X (gfx1250) — compile-verified
//
#include <hip/hip_runtime.h>

// ---------------------------------------------------------------------------
// CGConv x2 + BN + mean-pool + MLP head for MI455X (gfx1250, wave32, WMMA)
// Edge GEMMs (z[E,144] @ W[144,64]) run on V_WMMA_F32_16X16X4_F32.
// Gathered z tiles are staged with GLOBAL_LOAD_ASYNC_TO_LDS_B128 (ASYNCcnt).
// ---------------------------------------------------------------------------

typedef float v2f __attribute__((ext_vector_type(2)));
typedef float v8f __attribute__((ext_vector_type(8)));

#define D 64
#define ZDIM 144
#define ZPAD 148           // LDS row stride (floats); 148*4B -> 16 distinct banks
#define KSTEPS 36          // 144 / 4
#define PACKN ((ZDIM / 2) * D)   // float2 elements per packed weight matrix
#define WAVES_PER_BLOCK 4
#define EDGE_BLOCK (WAVES_PER_BLOCK * 32)
#define BN_EPS 1e-5f

// Branchless gate: sigmoid(f) * softplus(s)
//   sigmoid via v_exp + v_rcp; softplus(s) = max(s,0) + log(1+exp(-|s|))
//   (stable for all s: exp(-|s|) <= 1, so the log argument is in [1,2])
__device__ __forceinline__ float gate_msg(float f, float s) {
    const float sig = __builtin_amdgcn_rcpf(1.0f + __expf(-f));
    const float sp  = fmaxf(s, 0.0f) + __logf(1.0f + __expf(-fabsf(s)));
    return sig * sp;
}
__device__ __forceinline__ void atomic_add_f32(float* p, float v) {
    __hip_atomic_fetch_add(p, v, __ATOMIC_RELAXED, __HIP_MEMORY_SCOPE_AGENT);
}

// Async global -> LDS copy of 16 bytes per lane (CDNA5, tracked by ASYNCcnt).
// Flat shared pointers carry the wave-relative LDS offset in their low 32 bits.
__device__ __forceinline__ void async_b128(float* lds_ptr, const float* gptr) {
    unsigned loff = (unsigned)(uintptr_t)lds_ptr;
    asm volatile("global_load_async_to_lds_b128 %0, %1, off"
                 :: "v"(loff), "v"(gptr) : "memory");
}
__device__ __forceinline__ void wait_asynccnt0() {
    asm volatile("s_wait_asynccnt 0x0" ::: "memory");
}

// ---------------------------------------------------------------------------
// Weight packing: P[kp*64 + n] = (W[2kp][n], W[2kp+1][n])  -> one b64/fragment
// ---------------------------------------------------------------------------
__global__ void pack_weight_kernel(const float* __restrict__ W, v2f* __restrict__ P)
{
    const int idx = blockIdx.x * blockDim.x + threadIdx.x;
    if (idx >= PACKN) return;
    const int kp = idx >> 6;
    const int n  = idx & 63;
    v2f p;
    p[0] = W[(2 * kp) * D + n];
    p[1] = W[(2 * kp + 1) * D + n];
    P[idx] = p;
}

// ---------------------------------------------------------------------------
// Edge kernel: per 16-edge tile, build z = [x[dst] | x[src] | e_attr] in LDS
// via async copies, run two 16x144 @ 144x64 GEMMs on WMMA f32 16x16x4,
// gate with sigmoid*softplus, scatter-add into agg[dst].
// ---------------------------------------------------------------------------
__global__ void __launch_bounds__(EDGE_BLOCK)
cgconv_edge_kernel(const float* __restrict__ x,
                   const int*   __restrict__ ei,   // [2, E] (row0=src j, row1=dst i)
                   const float* __restrict__ ea,   // [E, 16]
                   const v2f* __restrict__ WfP, const float* __restrict__ bf,
                   const v2f* __restrict__ WsP, const float* __restrict__ bs,
                   float* __restrict__ agg,        // [N, 64], pre-zeroed
                   int nEdges, int nTiles, int iters)
{
    __shared__ float zbuf[WAVES_PER_BLOCK][16 * ZPAD];
    __shared__ int   dstbuf[WAVES_PER_BLOCK][16];

    const int lane  = threadIdx.x & 31;
    const int wave  = threadIdx.x >> 5;
    const int half  = lane >> 4;    // 0: lanes 0-15, 1: lanes 16-31
    const int l16   = lane & 15;
    const int gwave = blockIdx.x * WAVES_PER_BLOCK + wave;
    const int nwaves = gridDim.x * WAVES_PER_BLOCK;

    float* zw = &zbuf[wave][0];
    int*   dw = &dstbuf[wave][0];
    const int n0 = l16;

    // tile-invariant per-lane biases (one column per lane per N-tile)
    float bfv[4], bsv[4];
    #pragma unroll
    for (int nt = 0; nt < 4; ++nt) {
        bfv[nt] = bf[nt * 16 + n0];
        bsv[nt] = bs[nt * 16 + n0];
    }

    for (int it = 0; it < iters; ++it) {
        const int tile = gwave + it * nwaves;
        const bool active = (tile < nTiles);   // wave-uniform

        __syncthreads();
        if (active) {
            // ---- stage z tile: row r = edge tile*16+r; cols [dst x | src x | ea]
            const int e  = tile * 16 + l16;
            const bool ev = (e < nEdges);
            if (half == 0) {
                int di = ev ? ei[nEdges + e] : -1;
                dw[l16] = di;
                float* zr = zw + l16 * ZPAD;
                if (ev) {
                    const float* xr = x + (size_t)di * D;
                    #pragma unroll
                    for (int j = 0; j < 16; ++j) async_b128(zr + j * 4, xr + j * 4);
                    const float* er = ea + (size_t)e * 16;
                    #pragma unroll
                    for (int j = 0; j < 4; ++j) async_b128(zr + 128 + j * 4, er + j * 4);
                } else {
                    #pragma unroll
                    for (int j = 0; j < 64; ++j) zr[j] = 0.0f;
                    #pragma unroll
                    for (int j = 0; j < 16; ++j) zr[128 + j] = 0.0f;
                }
            } else {
                float* zr = zw + l16 * ZPAD + 64;
                if (ev) {
                    const int si = ei[e];
                    const float* xr = x + (size_t)si * D;
                    #pragma unroll
                    for (int j = 0; j < 16; ++j) async_b128(zr + j * 4, xr + j * 4);
                } else {
                    #pragma unroll
                    for (int j = 0; j < 64; ++j) zr[j] = 0.0f;
                }
            }
        }
        wait_asynccnt0();
        __syncthreads();
        if (!active) continue;

        // ---- GEMM: D = A(16x144) x B(144x64), K in steps of 4
        v8f accF[4], accS[4];
        #pragma unroll
        for (int nt = 0; nt < 4; ++nt)
            #pragma unroll
            for (int j = 0; j < 8; ++j) { accF[nt][j] = 0.0f; accS[nt][j] = 0.0f; }

        const float* zrow = zw + l16 * ZPAD;
        const int koff = half * 2;   // lanes 0-15 hold K={0,1}, lanes 16-31 K={2,3}

        #pragma unroll 2
        for (int k = 0; k < KSTEPS; ++k) {
            const int kb = 4 * k + koff;
            const v2f a = *(const v2f*)(zrow + kb);            // ds_load_b64
            const v2f* wf = WfP + (size_t)(2 * k + half) * D;
            const v2f* ws = WsP + (size_t)(2 * k + half) * D;
            #pragma unroll
            for (int nt = 0; nt < 4; ++nt) {
                const int n = nt * 16 + n0;
                const v2f bF = wf[n];                          // global_load_b64
                const v2f bS = ws[n];
                accF[nt] = __builtin_amdgcn_wmma_f32_16x16x4_f32(
                    false, a, false, bF, (short)0, accF[nt], false, false);
                accS[nt] = __builtin_amdgcn_wmma_f32_16x16x4_f32(
                    false, a, false, bS, (short)0, accS[nt], false, false);
            }
        }

        // ---- bias + sigmoid*softplus gate + scatter-add into agg[dst]
        // dst indices for this lane's 8 rows (contiguous in LDS -> b128 loads)
        int dloc[8];
        #pragma unroll
        for (int j = 0; j < 8; ++j) dloc[j] = dw[half * 8 + j];

        const bool fullTile = (tile * 16 + 16 <= nEdges);  // wave-uniform
        if (fullTile) {
            #pragma unroll
            for (int j = 0; j < 8; ++j) {
                float* arow = agg + (size_t)dloc[j] * D + n0;
                #pragma unroll
                for (int nt = 0; nt < 4; ++nt) {
                    const float msg = gate_msg(accF[nt][j] + bfv[nt],
                                               accS[nt][j] + bsv[nt]);
                    atomic_add_f32(arow + nt * 16, msg);
                }
            }
        } else {
            #pragma unroll
            for (int j = 0; j < 8; ++j) {
                if (dloc[j] >= 0) {
                    float* arow = agg + (size_t)dloc[j] * D + n0;
                    #pragma unroll
                    for (int nt = 0; nt < 4; ++nt) {
                        const float msg = gate_msg(accF[nt][j] + bfv[nt],
                                                   accS[nt][j] + bsv[nt]);
                        atomic_add_f32(arow + nt * 16, msg);
                    }
                }
            }
        }
    }
}

// ---------------------------------------------------------------------------
// BatchNorm pass 1: per-channel sum / sumsq of (x + agg)
// ---------------------------------------------------------------------------
__global__ void bn_reduce_kernel(const float* __restrict__ xin,
                                 const float* __restrict__ agg,
                                 float* __restrict__ stats,  // [128] pre-zeroed
                                 int nNodes)
{
    __shared__ float sh0[256];
    __shared__ float sh1[256];
    const int c   = threadIdx.x & 63;
    const int sub = threadIdx.x >> 6;            // 0..3
    float s = 0.0f, s2 = 0.0f;
    for (int i = blockIdx.x * 4 + sub; i < nNodes; i += gridDim.x * 4) {
        const size_t idx = (size_t)i * D + c;
        const float h = xin[idx] + agg[idx];
        s += h; s2 += h * h;
    }
    sh0[threadIdx.x] = s;
    sh1[threadIdx.x] = s2;
    __syncthreads();
    if (threadIdx.x < 64) {
        float ts = sh0[c] + sh0[c + 64] + sh0[c + 128] + sh0[c + 192];
        float t2 = sh1[c] + sh1[c + 64] + sh1[c + 128] + sh1[c + 192];
        atomic_add_f32(&stats[c], ts);
        atomic_add_f32(&stats[64 + c], t2);
    }
}

// ---------------------------------------------------------------------------
// BatchNorm pass 2: h = ((x+agg) - mu) * rsqrt(var+eps) * gamma + beta
// ---------------------------------------------------------------------------
__global__ void bn_apply_kernel(const float* __restrict__ xin,
                                const float* __restrict__ agg,
                                const float* __restrict__ stats,
                                const float* __restrict__ gamma,
                                const float* __restrict__ beta,
                                float* __restrict__ hout, int nNodes)
{
    const size_t idx = (size_t)blockIdx.x * blockDim.x + threadIdx.x;
    const size_t total = (size_t)nNodes * D;
    if (idx >= total) return;
    const int c = (int)(idx & 63);
    const float invN = 1.0f / (float)nNodes;
    const float mu   = stats[c] * invN;
    const float var  = stats[64 + c] * invN - mu * mu;
    const float scale = rsqrtf(var + BN_EPS) * gamma[c];
    const float h = xin[idx] + agg[idx];
    hout[idx] = (h - mu) * scale + beta[c];
}

// ---------------------------------------------------------------------------
// Segment-sum pooling (+ counts)
// ---------------------------------------------------------------------------
__global__ void pool_accum_kernel(const float* __restrict__ h,
                                  const int* __restrict__ batch,
                                  float* __restrict__ pooled,  // [G,64] zeroed
                                  float* __restrict__ cnt,     // [G] zeroed
                                  int nNodes)
{
    const size_t idx = (size_t)blockIdx.x * blockDim.x + threadIdx.x;
    const size_t total = (size_t)nNodes * D;
    if (idx >= total) return;
    const int i = (int)(idx >> 6);
    const int c = (int)(idx & 63);
    const int g = batch[i];
    atomic_add_f32(&pooled[(size_t)g * D + c], h[idx]);
    if (c == 0) atomic_add_f32(&cnt[g], 1.0f);
}

// ---------------------------------------------------------------------------
// Mean + Linear -> LeakyReLU(0.01) -> Linear  (one 64-thread block per graph)
// ---------------------------------------------------------------------------
__global__ void proj_kernel(const float* __restrict__ pooled,
                            const float* __restrict__ cnt,
                            const float* __restrict__ pw1, const float* __restrict__ pb1,
                            const float* __restrict__ pw2, const float* __restrict__ pb2,
                            float* __restrict__ out, int nGraphs)
{
    __shared__ float p[64];
    __shared__ float y1[64];
    const int g = blockIdx.x;
    if (g >= nGraphs) return;
    const int c = threadIdx.x;
    float cv = cnt[g];
    cv = cv > 1.0f ? cv : 1.0f;
    p[c] = pooled[(size_t)g * D + c] / cv;
    __syncthreads();
    float a = pb1[c];
    #pragma unroll 8
    for (int k = 0; k < D; ++k) a += p[k] * pw1[k * D + c];
    a = (a > 0.0f) ? a : 0.01f * a;
    y1[c] = a;
    __syncthreads();
    float o = pb2[c];
    #pragma unroll 8
    for (int k = 0; k < D; ++k) o += y1[k] * pw2[k * D + c];
    out[(size_t)g * D + c] = o;
}

// ---------------------------------------------------------------------------
// Host launcher
// ---------------------------------------------------------------------------
extern "C" void kernel_launch(void* const* d_in, const int* in_sizes, int n_in,
                              void* d_out, int out_size, void* d_ws, size_t ws_size,
                              hipStream_t stream)
{
    const float* x    = (const float*)d_in[0];
    const int*   ei   = (const int*)  d_in[1];
    const float* ea   = (const float*)d_in[2];
    const int*   batch= (const int*)  d_in[3];
    const float* fw0  = (const float*)d_in[4];
    const float* fb0  = (const float*)d_in[5];
    const float* sw0  = (const float*)d_in[6];
    const float* sb0  = (const float*)d_in[7];
    const float* bng0 = (const float*)d_in[8];
    const float* bnb0 = (const float*)d_in[9];
    const float* fw1  = (const float*)d_in[10];
    const float* fb1  = (const float*)d_in[11];
    const float* sw1  = (const float*)d_in[12];
    const float* sb1  = (const float*)d_in[13];
    const float* bng1 = (const float*)d_in[14];
    const float* bnb1 = (const float*)d_in[15];
    const float* pw1  = (const float*)d_in[16];
    const float* pb1  = (const float*)d_in[17];
    const float* pw2  = (const float*)d_in[18];
    const float* pb2  = (const float*)d_in[19];
    float* out = (float*)d_out;

    const int nNodes  = in_sizes[0] / D;
    const int nEdges  = in_sizes[1] / 2;
    const int nGraphs = out_size / D;

    // workspace layout (floats):
    //   agg | h1 | h2 | stats(128) | pooled | cnt | packed weights (4 x 2*PACKN)
    float* agg    = (float*)d_ws;
    float* h1     = agg + (size_t)nNodes * D;
    float* h2     = h1  + (size_t)nNodes * D;
    float* stats  = h2  + (size_t)nNodes * D;
    float* pooled = stats + 128;
    float* cnt    = pooled + (size_t)nGraphs * D;
    v2f* fw0p = (v2f*)(cnt + nGraphs);
    v2f* sw0p = fw0p + PACKN;
    v2f* fw1p = sw0p + PACKN;
    v2f* sw1p = fw1p + PACKN;

    const int nTiles = (nEdges + 15) / 16;
    const int edgeBlocks = 1024;
    const int nwaves = edgeBlocks * WAVES_PER_BLOCK;
    const int iters = (nTiles + nwaves - 1) / nwaves;   // uniform trip count

    const size_t nodeElems = (size_t)nNodes * D;
    const int elemBlocks = (int)((nodeElems + 255) / 256);
    const int bnRedBlocks = 256;
    const int packBlocks = (PACKN + 255) / 256;

    // ---------------- pack weights (K-pair interleave -> b64 fragments) ----
    pack_weight_kernel<<<packBlocks, 256, 0, stream>>>(fw0, fw0p);
    pack_weight_kernel<<<packBlocks, 256, 0, stream>>>(sw0, sw0p);
    pack_weight_kernel<<<packBlocks, 256, 0, stream>>>(fw1, fw1p);
    pack_weight_kernel<<<packBlocks, 256, 0, stream>>>(sw1, sw1p);

    // ---------------- layer 0 ----------------
    hipMemsetAsync(agg, 0, nodeElems * sizeof(float), stream);
    cgconv_edge_kernel<<<edgeBlocks, EDGE_BLOCK, 0, stream>>>(
        x, ei, ea, fw0p, fb0, sw0p, sb0, agg, nEdges, nTiles, iters);
    hipMemsetAsync(stats, 0, 128 * sizeof(float), stream);
    bn_reduce_kernel<<<bnRedBlocks, 256, 0, stream>>>(x, agg, stats, nNodes);
    bn_apply_kernel<<<elemBlocks, 256, 0, stream>>>(x, agg, stats, bng0, bnb0, h1, nNodes);

    // ---------------- layer 1 ----------------
    hipMemsetAsync(agg, 0, nodeElems * sizeof(float), stream);
    cgconv_edge_kernel<<<edgeBlocks, EDGE_BLOCK, 0, stream>>>(
        h1, ei, ea, fw1p, fb1, sw1p, sb1, agg, nEdges, nTiles, iters);
    hipMemsetAsync(stats, 0, 128 * sizeof(float), stream);
    bn_reduce_kernel<<<bnRedBlocks, 256, 0, stream>>>(h1, agg, stats, nNodes);
    bn_apply_kernel<<<elemBlocks, 256, 0, stream>>>(h1, agg, stats, bng1, bnb1, h2, nNodes);

    // ---------------- pool + head ----------------
    hipMemsetAsync(pooled, 0, (size_t)nGraphs * D * sizeof(float), stream);
    hipMemsetAsync(cnt, 0, (size_t)nGraphs * sizeof(float), stream);
    pool_accum_kernel<<<elemBlocks, 256, 0, stream>>>(h2, batch, pooled, cnt, nNodes);
    proj_kernel<<<nGraphs, 64, 0, stream>>>(pooled, cnt, pw1, pb1, pw2, pb2, out, nGraphs);
}